// TorchSumLayer_26723286515900
// MI455X (gfx1250) — compile-verified
//
#include <hip/hip_runtime.h>

// Problem dims (fixed by the reference).
#define B_DIM 4096
#define C_DIM 2048
#define N_DIM 2048
#define K_DIM 16

// Tiling: each block stages TB contiguous batch rows of x (TB*C*4 = 64 KB)
// into LDS via async copies, and computes TN nodes (one per thread).
#define TB 8
#define TN 256
#define NT 256  // threads per block (8 wave32s)

__global__ __launch_bounds__(NT) void spn_sumlayer_kernel(
    const float* __restrict__ x,     // [B, C]
    const float* __restrict__ logw,  // [N, K]
    const int*   __restrict__ idxs,  // [N, K]
    float*       __restrict__ out)   // [B, N]
{
    __shared__ __align__(16) float xs[TB * C_DIM];  // 64 KB

    const int tid = threadIdx.x;
    const int n   = blockIdx.x * TN + tid;   // node handled by this thread
    const int b0  = blockIdx.y * TB;         // first batch row of this tile

    // ---- Kick off async staging of 8 x-rows (contiguous 64 KB) into LDS.
    //      Each lane moves 16 B per instruction; 16 instructions/thread.
    {
        const unsigned long long gbase =
            (unsigned long long)(const void*)(x + (size_t)b0 * C_DIM);
        const uint32_t lds0 = (uint32_t)(uintptr_t)xs;
        #pragma unroll
        for (int i = 0; i < (TB * C_DIM * 4) / (16 * NT); ++i) {
            const int seg = tid + i * NT;
            const uint32_t lds_addr = lds0 + (uint32_t)seg * 16u;
            const unsigned long long gaddr = gbase + (unsigned long long)seg * 16ull;
            asm volatile("global_load_async_to_lds_b128 %0, %1, off"
                         :: "v"(lds_addr), "v"(gaddr)
                         : "memory");
        }
    }

    // ---- While the async copy is in flight (ASYNCcnt), pull this thread's
    //      node indices & log-weights into VGPRs (LOADcnt — independent).
    int   idx[K_DIM];
    float lw[K_DIM];
    {
        const int4*   ip = (const int4*)(idxs + (size_t)n * K_DIM);
        const float4* wp = (const float4*)(logw + (size_t)n * K_DIM);
        #pragma unroll
        for (int q = 0; q < K_DIM / 4; ++q) {
            const int4   iv = ip[q];
            const float4 wv = wp[q];
            idx[q * 4 + 0] = iv.x; idx[q * 4 + 1] = iv.y;
            idx[q * 4 + 2] = iv.z; idx[q * 4 + 3] = iv.w;
            lw[q * 4 + 0] = wv.x;  lw[q * 4 + 1] = wv.y;
            lw[q * 4 + 2] = wv.z;  lw[q * 4 + 3] = wv.w;
        }
    }

    // ---- Wait for our wave's async copies, then barrier so all waves see LDS.
    asm volatile("s_wait_asynccnt 0" ::: "memory");
    __syncthreads();

    const float LOG2E = 1.4426950408889634f;
    const float LN2   = 0.6931471805599453f;

    // ---- Gather from LDS + weighted logsumexp, one output per (b, n).
    //      Full unroll lets the compiler pair gathers across adjacent batch
    //      rows into ds_load_2addr_stride64_b32 (rows are 8192 B apart).
    #pragma unroll
    for (int bl = 0; bl < TB; ++bl) {
        const float* row = xs + bl * C_DIM;

        float y[K_DIM];
        #pragma unroll
        for (int k = 0; k < K_DIM; ++k)
            y[k] = row[idx[k]] + lw[k];        // LDS gathers

        float m = y[0];
        #pragma unroll
        for (int k = 1; k < K_DIM; ++k)
            m = fmaxf(m, y[k]);

        float s = 0.0f;
        #pragma unroll
        for (int k = 0; k < K_DIM; ++k)
            s += __builtin_amdgcn_exp2f((y[k] - m) * LOG2E);  // v_exp_f32

        // log(s) = log2(s) * ln2 ; v_log_f32 is base-2.
        const float r = m + __builtin_amdgcn_logf(s) * LN2;
        __builtin_nontemporal_store(r, &out[(size_t)(b0 + bl) * N_DIM + n]);
    }
}

extern "C" void kernel_launch(void* const* d_in, const int* in_sizes, int n_in,
                              void* d_out, int out_size, void* d_ws, size_t ws_size,
                              hipStream_t stream) {
    (void)in_sizes; (void)n_in; (void)out_size; (void)d_ws; (void)ws_size;

    const float* x    = (const float*)d_in[0];  // [B, C] f32
    const float* logw = (const float*)d_in[1];  // [N, K] f32
    const int*   idxs = (const int*)d_in[2];    // [N, K] i32
    float*       out  = (float*)d_out;          // [B, N] f32

    dim3 grid(N_DIM / TN, B_DIM / TB);  // (8, 512) blocks
    spn_sumlayer_kernel<<<grid, NT, 0, stream>>>(x, logw, idxs, out);
}